// Mask_proteins_38388417691667
// MI455X (gfx1250) — compile-verified
//
#include <hip/hip_runtime.h>
#include <cstdint>
#include <cstddef>

// ============================================================================
// Mask_proteins on MI455X (gfx1250), wave32.
//
// out[b,p,n] = ((x[b,p]-mean[p])/std[p]) * alpha[p,n]
//            + (1 - alpha[p,n]/max_a[n]) * 0.5 * z(b,p,n)
// alpha[p,n] = 25 * w[i(p),n] * w[j(p),n]   (triu pair, k=4)
// z = jax normal(key=42): threefry2x32 -> uniform(-1,1) -> sqrt(2)*erfinv
//
// Roofline: 126MB read + 505MB write ~= 631MB -> ~27us @ 23.3TB/s, but the
// PRNG (~45 VALU ops/elt, 5.7G ops total) dominates. JAX's threefry counter
// split pairs flat index m with m+HALF = (b+128, same p, same n), so each
// thread owns both halves of one threefry block (PRNG work halved).
//
// WMMA: per wave, V_WMMA_F32_16X16X4_F32 computes a 16(batch) x 16(4p x 4n)
// tile: A = xn (16x4), B = block-diagonal alpha, C = noise term, so the
// D = A*B + C accumulate fuses mask-multiply + noise-add.
// x is staged global->LDS with gfx1250 async-LDS loads (ASYNCcnt), read back
// transposed from padded LDS (stride 34: conflict-free, 8B aligned).
// ============================================================================

namespace {
constexpr int      kP      = 123256;                 // pairs (= INPUT_SIZE)
constexpr uint32_t kRow4P  = 4u * (uint32_t)kP;      // out floats per batch row
constexpr uint32_t kHalf   = 128u * kRow4P;          // threefry counter split
constexpr int      kPTiles = (kP + 3) / 4;           // 30814 wave tiles (4 p each)
constexpr int      kGridX  = (kPTiles + 7) / 8;      // 8 waves / block -> 3852
} // namespace

typedef __attribute__((ext_vector_type(2))) float v2f;
typedef __attribute__((ext_vector_type(8))) float v8f;

#if defined(__AMDGCN__)
  #if __has_builtin(__builtin_amdgcn_wmma_f32_16x16x4_f32)
    #define HAVE_WMMA_F32X4 1
  #endif
  #if __has_builtin(__builtin_amdgcn_global_load_async_to_lds_b32)
    #define HAVE_ASYNC_LDS 1
    typedef __attribute__((address_space(1))) int as1_int;
    typedef __attribute__((address_space(3))) int as3_int;
  #endif
#endif

// ---------------- threefry2x32, key = jax.random.key(42) = (0, 42) ----------
__device__ __forceinline__ uint32_t rotl32(uint32_t x, int r) {
  return (x << r) | (x >> (32 - r));
}
__device__ __forceinline__ void tf4(uint32_t& x0, uint32_t& x1,
                                    int a, int b, int c, int d) {
  x0 += x1; x1 = rotl32(x1, a); x1 ^= x0;
  x0 += x1; x1 = rotl32(x1, b); x1 ^= x0;
  x0 += x1; x1 = rotl32(x1, c); x1 ^= x0;
  x0 += x1; x1 = rotl32(x1, d); x1 ^= x0;
}
__device__ __forceinline__ uint2 threefry2x32(uint32_t c0, uint32_t c1) {
  const uint32_t ks0 = 0u, ks1 = 42u, ks2 = 0x1BD11BDAu ^ 0u ^ 42u;
  uint32_t x0 = c0 + ks0, x1 = c1 + ks1;
  tf4(x0, x1, 13, 15, 26, 6);  x0 += ks1; x1 += ks2 + 1u;
  tf4(x0, x1, 17, 29, 16, 24); x0 += ks2; x1 += ks0 + 2u;
  tf4(x0, x1, 13, 15, 26, 6);  x0 += ks0; x1 += ks1 + 3u;
  tf4(x0, x1, 17, 29, 16, 24); x0 += ks1; x1 += ks2 + 4u;
  tf4(x0, x1, 13, 15, 26, 6);  x0 += ks2; x1 += ks0 + 5u;
  return make_uint2(x0, x1);
}

// bits -> uniform(-1,1) -> sqrt(2)*erfinv (Giles single-precision approx)
__device__ __forceinline__ float bits_to_normal(uint32_t b) {
  float f = __uint_as_float(0x3f800000u | (b >> 9)) - 1.0f;   // [0,1)
  float x = fmaf(f, 1.99999994f, -0.99999994f);               // (-1,1)
  float w = -__logf(fmaf(-x, x, 1.0f));                       // -log(1-x^2)
  float p;
  if (w < 5.0f) {
    w -= 2.5f;
    p = 2.81022636e-08f;
    p = fmaf(p, w, 3.43273939e-07f);
    p = fmaf(p, w, -3.5233877e-06f);
    p = fmaf(p, w, -4.39150654e-06f);
    p = fmaf(p, w, 0.00021858087f);
    p = fmaf(p, w, -0.00125372503f);
    p = fmaf(p, w, -0.00417768164f);
    p = fmaf(p, w, 0.246640727f);
    p = fmaf(p, w, 1.50140941f);
  } else {
    w = sqrtf(w) - 3.0f;
    p = -0.000200214257f;
    p = fmaf(p, w, 0.000100950558f);
    p = fmaf(p, w, 0.00134934322f);
    p = fmaf(p, w, -0.00367342844f);
    p = fmaf(p, w, 0.00573950773f);
    p = fmaf(p, w, -0.0076224613f);
    p = fmaf(p, w, 0.00943887047f);
    p = fmaf(p, w, 1.00167406f);
    p = fmaf(p, w, 2.83297682f);
  }
  return 1.41421356f * p * x;
}

// ---------------- kernel 1: residue weights + max_a into ws -----------------
// ws[0 .. 1999]  : w[res][n]  (500 x 4)
// ws[2000..2003] : max_a[n] = 25 * max_pairs(w_i * w_j)
__global__ __launch_bounds__(256) void setup_kernel(
    const float* __restrict__ weights, float* __restrict__ ws) {
  __shared__ float psm[4][52];
  __shared__ float wn[500][4];
  __shared__ float smax[4][500];
  const int t = threadIdx.x;

  if (t < 4) {                       // softmax(weights, axis=1) * 52
    float mx = -3.4e38f;
    for (int k = 0; k < 52; ++k) mx = fmaxf(mx, weights[t * 52 + k]);
    float s = 0.0f;
    for (int k = 0; k < 52; ++k) {
      float e = __expf(weights[t * 52 + k] - mx);
      psm[t][k] = e; s += e;
    }
    float inv = 52.0f / s;
    for (int k = 0; k < 52; ++k) psm[t][k] *= inv;
  }
  __syncthreads();

  for (int i = t; i < 500; i += 256) {   // product of 3 bins, relu, normalize
    int b = i / 10;                      // BALANCE = 0, SKIP = 10
    float r[4]; float sum = 2.5f;        // relu(FACTOR_FAKE - CUTOFF) = 2.5
    for (int n = 0; n < 4; ++n) {
      float v = psm[n][b] * psm[n][b + 1] * psm[n][b + 2];
      v = fmaxf(v - 0.5f, 0.0f);
      r[n] = v; sum += v;
    }
    float invs = 1.0f / sum;
    for (int n = 0; n < 4; ++n) {
      float vv = r[n] * invs;
      wn[i][n] = vv;
      ws[i * 4 + n] = vv;
    }
  }
  __syncthreads();

  if (t < 4) {                       // max over valid pairs via suffix max
    const int n = t;
    float run = wn[499][n]; smax[n][499] = run;
    for (int i = 498; i >= 0; --i) { run = fmaxf(run, wn[i][n]); smax[n][i] = run; }
    float best = 0.0f;
    for (int i = 0; i <= 495; ++i) best = fmaxf(best, wn[i][n] * smax[n][i + 4]);
    ws[2000 + n] = 25.0f * best;
  }
}

// ---------------- kernel 2: the big fused mask+noise kernel -----------------
// grid = (kGridX, 8); block = 256 (8 waves). Wave tile: 4 p * (16+16) batches.
__global__ __launch_bounds__(256) void mask_main(
    const float* __restrict__ x, const float* __restrict__ mean,
    const float* __restrict__ stdv, const float* __restrict__ wv,
    float* __restrict__ out) {
  __shared__ __align__(16) float tile[32 * 34];    // 32 batches x 32 p, pad->34
  const int t     = (int)threadIdx.x;
  const int p0blk = (int)blockIdx.x * 32;
  const int b0    = (int)blockIdx.y * 16;          // 0,16,...,112

  // ---- stage x tile: rows 0..15 = batches b0+r, rows 16..31 = b0+128+(r-16)
#if defined(HAVE_ASYNC_LDS)
  #pragma unroll
  for (int s = 0; s < 4; ++s) {
    int flat = t + 256 * s;
    int row = flat >> 5, col = flat & 31;
    int batch = (row < 16) ? (b0 + row) : (b0 + 112 + row);
    int gp = p0blk + col; gp = (gp < kP) ? gp : (kP - 1);
    const float* gsrc = x + (size_t)batch * kP + gp;
    __builtin_amdgcn_global_load_async_to_lds_b32(
        (as1_int*)gsrc, (as3_int*)&tile[row * 34 + col], 0, 0);
  }
  #if __has_builtin(__builtin_amdgcn_s_wait_asynccnt)
  __builtin_amdgcn_s_wait_asynccnt(0);
  #else
  asm volatile("s_wait_asynccnt 0" ::: "memory");
  #endif
  asm volatile("" ::: "memory");
  #pragma unroll
  for (int s = 0; s < 4; ++s) {      // normalize in place (own elements only)
    int flat = t + 256 * s;
    int row = flat >> 5, col = flat & 31;
    int gp = p0blk + col; gp = (gp < kP) ? gp : (kP - 1);
    int loff = row * 34 + col;
    tile[loff] = (tile[loff] - mean[gp]) / stdv[gp];
  }
#else
  #pragma unroll
  for (int s = 0; s < 4; ++s) {
    int flat = t + 256 * s;
    int row = flat >> 5, col = flat & 31;
    int batch = (row < 16) ? (b0 + row) : (b0 + 112 + row);
    int gp = p0blk + col; gp = (gp < kP) ? gp : (kP - 1);
    tile[row * 34 + col] = (x[(size_t)batch * kP + gp] - mean[gp]) / stdv[gp];
  }
#endif
  __syncthreads();

  const int wave = t >> 5, lane = t & 31;
  int p0w = ((int)blockIdx.x * 8 + wave) * 4;
  if (p0w > kP - 4) p0w = kP - 4;    // clamp: duplicate identical stores, OK
  const int hi   = lane >> 4;        // half-wave
  const int m    = lane & 15;        // D column / A row
  const int myPl = m >> 2;           // which of the 4 p's
  const int myN  = lane & 3;         // subsystem

  // ---- alpha/beta for this lane's (p, n): invert triangular pair index ----
  const int p = p0w + myPl;
  int i = (int)((993.0f - sqrtf((float)(986049 - 8 * p))) * 0.5f);
  i = (i < 0) ? 0 : ((i > 495) ? 495 : i);
  while (i > 0 && (i * (993 - i)) / 2 > p) --i;
  while (i < 495 && ((i + 1) * (993 - (i + 1))) / 2 <= p) ++i;
  const int j = p - (i * (993 - i)) / 2 + i + 4;
  const float aval = 25.0f * wv[i * 4 + myN] * wv[j * 4 + myN];
  const float maxa = wv[2000 + myN];
  const float beta = (1.0f - aval / maxa) * 0.5f;   // NOISE = 0.5

  // ---- A operands (16x4 f32: VGPR0 = K{0|2}, VGPR1 = K{1|3}) ----
  const int cb = (p0w - p0blk) + 2 * hi;            // even -> 8B aligned
  const v2f aA = *reinterpret_cast<const v2f*>(&tile[m * 34 + cb]);
  const v2f aB = *reinterpret_cast<const v2f*>(&tile[(16 + m) * 34 + cb]);

  // ---- B operand: block-diagonal alpha (nonzero iff col/4 == K) ----
  v2f bm;
  bm.x = (myPl == 2 * hi)     ? aval : 0.0f;        // K = 0 | 2
  bm.y = (myPl == 2 * hi + 1) ? aval : 0.0f;        // K = 1 | 3

  // ---- C operand: noise term; one threefry block covers (b, b+128) ----
  v8f cA, cB;
  const uint32_t colOff = (uint32_t)p0w * 4u + (uint32_t)m;
  #pragma unroll
  for (int v = 0; v < 8; ++v) {
    uint32_t bA  = (uint32_t)(b0 + v + 8 * hi);     // 0..127
    uint32_t idx = bA * kRow4P + colOff;            // flat (b,p,n) index
    uint2 r = threefry2x32(idx, idx + kHalf);
    cA[v] = beta * bits_to_normal(r.x);
    cB[v] = beta * bits_to_normal(r.y);
  }

  // ---- D = A x B + C : fused mask-multiply + noise-add ----
#if defined(HAVE_WMMA_F32X4)
  v8f dA = __builtin_amdgcn_wmma_f32_16x16x4_f32(false, aA, false, bm,
                                                 (short)0, cA, false, false);
  v8f dB = __builtin_amdgcn_wmma_f32_16x16x4_f32(false, aB, false, bm,
                                                 (short)0, cB, false, false);
#else
  v8f dA, dB;
  const int cfix = (p0w - p0blk) + myPl;
  #pragma unroll
  for (int v = 0; v < 8; ++v) {
    dA[v] = fmaf(tile[(v + 8 * hi) * 34 + cfix], aval, cA[v]);
    dB[v] = fmaf(tile[(16 + v + 8 * hi) * 34 + cfix], aval, cB[v]);
  }
  (void)aA; (void)aB; (void)bm;
#endif

  // ---- stores: 16 consecutive floats per half-wave per VGPR; streaming ----
  float* op = out + (size_t)p0w * 4 + m;
  #pragma unroll
  for (int v = 0; v < 8; ++v) {
    size_t offA = (size_t)(b0 + v + 8 * hi) * kRow4P;
    __builtin_nontemporal_store(dA[v], op + offA);
    __builtin_nontemporal_store(dB[v], op + offA + (size_t)128 * kRow4P);
  }
}

// ---------------------------------------------------------------------------
extern "C" void kernel_launch(void* const* d_in, const int* in_sizes, int n_in,
                              void* d_out, int out_size, void* d_ws,
                              size_t ws_size, hipStream_t stream) {
  (void)in_sizes; (void)n_in; (void)out_size; (void)ws_size;
  const float* x       = (const float*)d_in[0];
  const float* mean    = (const float*)d_in[1];
  const float* stdv    = (const float*)d_in[2];
  const float* weights = (const float*)d_in[3];
  float* out = (float*)d_out;
  float* ws  = (float*)d_ws;          // needs 2004 floats (~8KB)

  setup_kernel<<<1, 256, 0, stream>>>(weights, ws);
  dim3 grid(kGridX, 8, 1);            // (3852 p-tiles, 8 batch pair-tiles)
  mask_main<<<grid, 256, 0, stream>>>(x, mean, stdv, ws, out);
}